// GCN_20521353741010
// MI455X (gfx1250) — compile-verified
//
#include <hip/hip_runtime.h>
#include <hip/hip_bf16.h>

#define N_NODES 100000
#define N_EDGES 3200000
#define IN_CH   128
#define HID     32

typedef __attribute__((ext_vector_type(2))) float v2f;
typedef __attribute__((ext_vector_type(4))) float v4f;
typedef __attribute__((ext_vector_type(8))) float v8f;

typedef long long idx_t;  // reference declares edge_index as int64

// ---------- degree / norm ----------
__global__ void k_init_deg(float* deg, int n) {
    int i = blockIdx.x * blockDim.x + threadIdx.x;
    if (i < n) deg[i] = 1.0f;  // self-loop contributes 1 to every node's degree
}

__global__ void k_deg_count(const idx_t* __restrict__ dst, float* deg, int e) {
    int i = blockIdx.x * blockDim.x + threadIdx.x;
    if (i < e) atomicAdd(&deg[(int)dst[i]], 1.0f);
}

__global__ void k_deg_rsqrt(float* deg, int n) {
    int i = blockIdx.x * blockDim.x + threadIdx.x;
    if (i < n) {
        float d = deg[i];
        deg[i] = (d > 0.0f) ? rsqrtf(d) : 0.0f;   // in place: deg -> deg^{-1/2}
    }
}

__global__ void k_zero(float* p, int n) {
    int i = blockIdx.x * blockDim.x + threadIdx.x;
    if (i < n) p[i] = 0.0f;
}

// ---------- WMMA GEMM: out[M x 32] = A[M x K] @ W[K x 32], fp32 exact ----------
// One wave computes a 16x32 tile via V_WMMA_F32_16X16X4_F32 (two 16x16 N-tiles).
// W is staged into LDS pre-swizzled so each lane's per-k-step B data is one
// 16B-aligned ds_load_b128 directly into even-aligned VGPR pairs (no movs).
//
// Swizzled layout: sWz[(k0/4)*128 + lane*4 + {0..3}] where for lane (h=lane>>4,
// lm=lane&15):  [0]=W[(k0+2h)*32+lm]     [1]=W[(k0+2h+1)*32+lm]      (N-tile 0)
//               [2]=W[(k0+2h)*32+16+lm]  [3]=W[(k0+2h+1)*32+16+lm]   (N-tile 1)
template <int K>
__global__ __launch_bounds__(256) void k_gemm_wmma(const float* __restrict__ A,
                                                   const float* __restrict__ W,
                                                   float* __restrict__ out,
                                                   int mtiles) {
    __shared__ __align__(16) float sWz[K * 32];
    for (int idx = threadIdx.x; idx < K * 32; idx += 256) {
        const int s    = idx >> 7;          // k-step (4 K-values per step)
        const int rem  = idx & 127;
        const int ln   = rem >> 2;          // lane this word serves
        const int j    = rem & 3;           // word within lane's 16B
        const int h    = ln >> 4;
        const int lm   = ln & 15;
        const int k    = 4 * s + 2 * h + (j & 1);
        const int n    = ((j & 2) << 3) + lm;   // +16 for N-tile 1
        sWz[idx] = W[k * 32 + n];
    }
    __syncthreads();

    const int wave = blockIdx.x * (256 / 32) + (threadIdx.x >> 5);
    if (wave >= mtiles) return;   // uniform per wave: EXEC stays all-ones inside

    const int lane = threadIdx.x & 31;
    const int half = lane >> 4;   // 0: holds K=0,1 ; 1: holds K=2,3  (A layout)
    const int lm   = lane & 15;   // A: M index / D: N index

    const float* arow = A + (size_t)(wave * 16 + lm) * K + 2 * half;

    v8f c0 = {0.f, 0.f, 0.f, 0.f, 0.f, 0.f, 0.f, 0.f};
    v8f c1 = {0.f, 0.f, 0.f, 0.f, 0.f, 0.f, 0.f, 0.f};

    for (int k0 = 0; k0 < K; k0 += 4) {
        // A fragment (16x4): lane<16 -> {k0, k0+1}, lane>=16 -> {k0+2, k0+3}
        v2f a = *(const v2f*)(arow + k0);
        // B fragments for both N-tiles: one ds_load_b128
        const v4f bq = *(const v4f*)&sWz[(k0 << 5) + lane * 4];  // (k0/4)*128
        v2f b0 = { bq.x, bq.y };
        v2f b1 = { bq.z, bq.w };
        c0 = __builtin_amdgcn_wmma_f32_16x16x4_f32(false, a, false, b0,
                                                   (short)0, c0, false, false);
        c1 = __builtin_amdgcn_wmma_f32_16x16x4_f32(false, a, false, b1,
                                                   (short)0, c1, false, false);
    }

    // D layout: vgpr r -> row (r + 8*half), col lm (tile0) / 16+lm (tile1)
    float* orow = out + (size_t)(wave * 16) * 32;
    #pragma unroll
    for (int r = 0; r < 8; ++r) {
        const int row = r + 8 * half;
        orow[row * 32 + lm]      = c0[r];
        orow[row * 32 + 16 + lm] = c1[r];
    }
}

// ---------- edge aggregation: lanes = channels, one edge per wave-row ----------
__global__ void k_edge_agg(const idx_t* __restrict__ src,
                           const idx_t* __restrict__ dst,
                           const float* __restrict__ h,
                           const float* __restrict__ dis,
                           float* agg, int e) {
    int t  = blockIdx.x * blockDim.x + threadIdx.x;
    int ei = t >> 5;
    int c  = t & 31;
    if (ei >= e) return;
    int s = (int)src[ei];
    int d = (int)dst[ei];
    float norm = dis[s] * dis[d];
    float v = h[(size_t)s * 32 + c] * norm;        // coalesced 128B gather per wave
    atomicAdd(&agg[(size_t)d * 32 + c], v);        // global_atomic_add_f32 (L2-resident)
}

// ---------- fused self-loop + bias + ReLU (in place on agg) ----------
__global__ void k_self_bias_relu(float* agg, const float* __restrict__ h,
                                 const float* __restrict__ dis,
                                 const float* __restrict__ b, int n) {
    int t = blockIdx.x * blockDim.x + threadIdx.x;
    if (t >= n * 32) return;
    int i = t >> 5;
    int c = t & 31;
    float di = dis[i];
    float v = agg[t] + h[t] * di * di + b[c];
    agg[t] = v > 0.0f ? v : 0.0f;
}

// ---------- final projection 32 -> 2 ----------
__global__ void k_proj(const float* __restrict__ h, const float* __restrict__ Wc,
                       const float* __restrict__ bc, float* __restrict__ out, int n) {
    int i = blockIdx.x * blockDim.x + threadIdx.x;
    if (i >= n) return;
    float a0 = bc[0], a1 = bc[1];
    const float* hr = h + (size_t)i * 32;
    #pragma unroll
    for (int k = 0; k < 32; ++k) {
        float v = hr[k];
        a0 += v * Wc[k * 2 + 0];
        a1 += v * Wc[k * 2 + 1];
    }
    out[i * 2 + 0] = a0;
    out[i * 2 + 1] = a1;
}

extern "C" void kernel_launch(void* const* d_in, const int* in_sizes, int n_in,
                              void* d_out, int out_size, void* d_ws, size_t ws_size,
                              hipStream_t stream) {
    const float* x   = (const float*)d_in[0];
    const idx_t* ei  = (const idx_t*)d_in[1];   // [2, E]: row0=src, row1=dst
    const float* W1  = (const float*)d_in[2];
    const float* b1  = (const float*)d_in[3];
    const float* W2  = (const float*)d_in[4];
    const float* b2  = (const float*)d_in[5];
    const float* Wc  = (const float*)d_in[6];
    const float* bc  = (const float*)d_in[7];
    float* out = (float*)d_out;

    const idx_t* src = ei;
    const idx_t* dst = ei + N_EDGES;

    // workspace layout (floats): dis[N] | bufA[N*32] | bufB[N*32] | bufC[N*32]
    float* dis  = (float*)d_ws;
    float* bufA = dis  + N_NODES;
    float* bufB = bufA + (size_t)N_NODES * 32;
    float* bufC = bufB + (size_t)N_NODES * 32;

    const int T = 256;
    const int gN    = (N_NODES + T - 1) / T;            // per-node
    const int gE    = (N_EDGES + T - 1) / T;            // per-edge
    const int gNC   = (N_NODES * 32 + T - 1) / T;       // per (node, channel)
    const int gEC   = (int)(((size_t)N_EDGES * 32 + T - 1) / T); // per (edge, channel)
    const int mtiles = N_NODES / 16;                    // 6250, exact
    const int gGemm  = (mtiles + 8 - 1) / 8;            // 8 waves per block

    // --- normalization: deg -> deg^{-1/2} ---
    k_init_deg <<<gN, T, 0, stream>>>(dis, N_NODES);
    k_deg_count<<<gE, T, 0, stream>>>(dst, dis, N_EDGES);
    k_deg_rsqrt<<<gN, T, 0, stream>>>(dis, N_NODES);

    // --- layer 1: h1 = x @ W1 (WMMA fp32), aggregate, +b1, ReLU ---
    k_gemm_wmma<IN_CH><<<gGemm, T, 0, stream>>>(x, W1, bufA, mtiles);
    k_zero     <<<gNC, T, 0, stream>>>(bufB, N_NODES * 32);
    k_edge_agg <<<gEC, T, 0, stream>>>(src, dst, bufA, dis, bufB, N_EDGES);
    k_self_bias_relu<<<gNC, T, 0, stream>>>(bufB, bufA, dis, b1, N_NODES);

    // --- layer 2: h2 = relu1 @ W2 (WMMA fp32), aggregate, +b2, ReLU ---
    k_gemm_wmma<HID><<<gGemm, T, 0, stream>>>(bufB, W2, bufA, mtiles);
    k_zero     <<<gNC, T, 0, stream>>>(bufC, N_NODES * 32);
    k_edge_agg <<<gEC, T, 0, stream>>>(src, dst, bufA, dis, bufC, N_EDGES);
    k_self_bias_relu<<<gNC, T, 0, stream>>>(bufC, bufA, dis, b2, N_NODES);

    // --- classifier head ---
    k_proj<<<gN, T, 0, stream>>>(bufC, Wc, bc, out, N_NODES);
}